// MCPBRNN_Generic_PETconstraint_Two_VariantOutputGate_BYPASSM1_10196252360852
// MI455X (gfx1250) — compile-verified
//
#include <hip/hip_runtime.h>

#define SPIN 8760
#define TRAIN 400000
#define U1_MAX 221.519f
#define ML 2.9086f
#define SL 1.898f

#define OUT_LEN 64          // output rows per chunk/thread
#define WARM 192            // warm-up rows (f^192 << fp32 eps)
#define SEQ (WARM + OUT_LEN) // 256 rows per thread
#define TROWS 16            // rows per TDM tile per lane
#define NTILES (SEQ / TROWS) // 16
#define TPB 256
#define WAVES_PER_BLOCK (TPB / 32)

#if __has_builtin(__builtin_amdgcn_tensor_load_to_lds)
#define HAVE_TDM 1
#else
#define HAVE_TDM 0
#endif

#if __has_builtin(__builtin_amdgcn_s_wait_tensorcnt)
#define TCNT_WAIT(n) __builtin_amdgcn_s_wait_tensorcnt(n)
#else
#define TCNT_WAIT(n)
#endif

typedef unsigned int v4u_t __attribute__((ext_vector_type(4)));
typedef int v8i_t __attribute__((ext_vector_type(8)));
typedef int v4i_t __attribute__((ext_vector_type(4)));

__device__ __forceinline__ float sigm(float x) {
    return 1.0f / (1.0f + __expf(-x));
}

#if HAVE_TDM
// Issue a 2D TDM tile load: tile_dim0 = 2*TROWS f32 elems (contiguous 16 rows
// of x for one chunk), tile_dim1 = n_valid lanes, dim0 stride = 2*OUT_LEN elems
// (stride between adjacent lanes' chunks). Descriptor per CDNA5 ISA sec 8.
__device__ __forceinline__ void tdm_issue_tile(unsigned int lds_byte_off,
                                               const float* gptr,
                                               unsigned int n_valid) {
    unsigned long long ga = (unsigned long long)(uintptr_t)gptr;
    v4u_t g0;
    g0.x = 1u;                                   // count=1, user-mode load
    g0.y = lds_byte_off;                         // lds_addr (bytes)
    g0.z = (unsigned int)(ga & 0xFFFFFFFFu);     // global_addr[31:0]
    g0.w = (unsigned int)((ga >> 32) & 0x01FFFFFFu) | 0x80000000u; // type=2

    unsigned int td0 = 2u * TROWS;   // tensor_dim0 (elements)
    unsigned int td1 = n_valid;      // tensor_dim1 (lanes)
    unsigned int tl0 = 2u * TROWS;   // tile_dim0
    unsigned int tl1 = n_valid;      // tile_dim1
    unsigned long long st0 = 2ull * OUT_LEN; // tensor_dim0_stride (elements)

    v8i_t g1;
    g1[0] = (int)(2u << 16);                                   // data_size=4B
    g1[1] = (int)((td0 & 0xFFFFu) << 16);                      // dim0[15:0]
    g1[2] = (int)(((td0 >> 16) & 0xFFFFu) | ((td1 & 0xFFFFu) << 16));
    g1[3] = (int)(((td1 >> 16) & 0xFFFFu) | ((tl0 & 0xFFFFu) << 16));
    g1[4] = (int)(tl1 & 0xFFFFu);                              // tile_dim2=0
    g1[5] = (int)(unsigned int)(st0 & 0xFFFFFFFFull);
    g1[6] = (int)(unsigned int)((st0 >> 32) & 0xFFFFull);
    g1[7] = 0;

    v4i_t g2 = {0, 0, 0, 0};
    v4i_t g3 = {0, 0, 0, 0};
    v8i_t gx = {0, 0, 0, 0, 0, 0, 0, 0}; // extended group (6-arg toolchain)
    __builtin_amdgcn_tensor_load_to_lds(g0, g1, g2, g3, gx, 0);
}
#endif

// ---------------- workspace reduction kernels ----------------

__global__ void ws_init_kernel(double* acc) {
    if (threadIdx.x == 0 && blockIdx.x == 0) {
        acc[0] = 0.0;  // sum
        acc[1] = 0.0;  // sum of squares
        ((float*)acc)[4] = 0.0f; // obsstd slot
    }
}

__global__ void stats_kernel(const float* __restrict__ y, int n0, int n1,
                             double* acc) {
    __shared__ double ssum[TPB];
    __shared__ double ssq[TPB];
    double s = 0.0, q = 0.0;
    for (int i = n0 + blockIdx.x * blockDim.x + threadIdx.x; i < n1;
         i += gridDim.x * blockDim.x) {
        double v = (double)y[i];
        s += v;
        q += v * v;
    }
    ssum[threadIdx.x] = s;
    ssq[threadIdx.x] = q;
    __syncthreads();
    for (int off = TPB / 2; off > 0; off >>= 1) {
        if (threadIdx.x < off) {
            ssum[threadIdx.x] += ssum[threadIdx.x + off];
            ssq[threadIdx.x] += ssq[threadIdx.x + off];
        }
        __syncthreads();
    }
    if (threadIdx.x == 0) {
        atomicAdd(&acc[0], ssum[0]);
        atomicAdd(&acc[1], ssq[0]);
    }
}

__global__ void finalize_kernel(double* acc, int n) {
    if (threadIdx.x == 0 && blockIdx.x == 0) {
        double s = acc[0], q = acc[1];
        double var = 0.0;
        if (n > 1) var = (q - s * s / (double)n) / (double)(n - 1);
        if (var < 0.0) var = 0.0;
        ((float*)acc)[4] = (float)sqrt(var);
    }
}

// zero rows r < time_lag across all 15 output columns
__global__ void zerofill_kernel(float* __restrict__ out,
                                const int* __restrict__ tl_p, int B) {
    int tl = tl_p[0];
    for (int r = blockIdx.x * blockDim.x + threadIdx.x; r < tl;
         r += gridDim.x * blockDim.x) {
        for (int a = 0; a < 12; ++a) out[(size_t)a * B + r] = 0.0f;
        out[(size_t)12 * B + 2 * r] = 0.0f;
        out[(size_t)12 * B + 2 * r + 1] = 0.0f;
        out[(size_t)14 * B + r] = 0.0f;
    }
}

// ---------------- main scan kernel ----------------

__global__ void __launch_bounds__(TPB)
mcpbrnn_scan_kernel(const float* __restrict__ x,
                    const int* __restrict__ tl_p,
                    const float* __restrict__ p_mean,
                    const float* __restrict__ p_std,
                    const float* __restrict__ w_r_yom,
                    const float* __restrict__ w_r_yom_gw,
                    const float* __restrict__ w_r_ylm,
                    const float* __restrict__ w_r_yfm,
                    const float* __restrict__ b0_yom,
                    const float* __restrict__ w1_yom,
                    const float* __restrict__ b0_gw,
                    const float* __restrict__ w1_gw,
                    const float* __restrict__ b0_ylm,
                    const float* __restrict__ w2_ylm,
                    const float* __restrict__ w1_yum,
                    const float* __restrict__ b0_yum,
                    const double* __restrict__ ws,
                    float* __restrict__ out, int B) {
    __shared__ float tile[WAVES_PER_BLOCK][2][32][TROWS][2]; // 64 KB

    const int time_lag = tl_p[0];
    const int N = B - time_lag;            // scan length
    if (N <= 0) return;
    const int chunkCount = (N + OUT_LEN - 1) / OUT_LEN;

    const int wv = __builtin_amdgcn_readfirstlane((int)(threadIdx.x >> 5));
    const int lane = (int)(threadIdx.x & 31);
    const int chunkBase = (int)blockIdx.x * TPB + wv * 32; // wave-uniform
    if (chunkBase >= chunkCount) return;                   // whole wave idle
    const int k = chunkBase + lane;                        // this lane's chunk
    const bool kValid = (k < chunkCount);

    int nValid = chunkCount - chunkBase;
    if (nValid > 32) nValid = 32;

    // TDM legal iff no lane reads before row time_lag or past row B
#if HAVE_TDM
    const bool useTDM =
        (chunkBase * OUT_LEN >= WARM) &&
        ((long long)(chunkBase + nValid) * OUT_LEN + time_lag <= (long long)B);
#else
    const bool useTDM = false;
#endif

    // scalar parameters (wave-uniform loads)
    const float mo = p_mean[0];
    const float inv_so = 1.0f / p_std[0];
    const float e_o = __expf(w_r_yom[0]);
    const float e_gw = __expf(w_r_yom_gw[0]);
    const float e_l = __expf(w_r_ylm[0]);
    const float e_f = __expf(w_r_yfm[0]);
    const float inv_den = 1.0f / (e_o + e_gw + e_l + e_f);
    const float oo1 = e_o * inv_den;
    const float oogw1 = e_gw * inv_den;
    const float ol1 = e_l * inv_den;
    const float wyum = w1_yum[0], byum = b0_yum[0];
    const float wyom = w1_yom[0], byom = b0_yom[0];
    const float wgw = w1_gw[0], bgw = b0_gw[0];
    const float wylm = w2_ylm[0], bylm = b0_ylm[0];
    const float wyum_u = wyum * (1.0f / U1_MAX);
    const float inv_SL = 1.0f / SL;
    const float stdv = ((const float*)ws)[4];

    const int jStartRaw = WARM - k * OUT_LEN;
    const int jStart = jStartRaw > 0 ? jStartRaw : 0; // chunk0: no warm-up
    int outLen = N - k * OUT_LEN;
    if (outLen > OUT_LEN) outLen = OUT_LEN;

    const int seqBase = chunkBase * OUT_LEN - WARM; // scan row of wave j=0,y=0

    // tile filler: TDM DMA (fast path) or per-lane clamped loads (edges)
    auto issue_tile = [&](int t, int buf) {
#if HAVE_TDM
        if (useTDM) {
            const float* g = x + 2ull * (unsigned long long)(time_lag + seqBase + t * TROWS);
            unsigned int lds_off =
                (unsigned int)(uintptr_t)(&tile[wv][buf][0][0][0]);
            tdm_issue_tile(lds_off, g, (unsigned int)nValid);
            return;
        }
#endif
        // fallback: each lane fills its own 16 rows, rows clamped in-bounds
        for (int r = 0; r < TROWS; ++r) {
            int srow = k * OUT_LEN - WARM + t * TROWS + r;
            if (srow < 0) srow = 0;
            if (srow >= N) srow = N - 1;
            const float* p = x + 2ull * (unsigned long long)(time_lag + srow);
            tile[wv][buf][lane][r][0] = p[0];
            tile[wv][buf][lane][r][1] = p[1];
        }
    };

    float c = 0.0f;
    int buf = 0;
    issue_tile(0, 0);

    for (int t = 0; t < NTILES; ++t) {
        const bool more = (t + 1 < NTILES);
        if (more) issue_tile(t + 1, buf ^ 1);
        if (useTDM) {
            if (more) { TCNT_WAIT(1); } else { TCNT_WAIT(0); }
        }
        asm volatile("" ::: "memory"); // LDS written by DMA: force reload

        for (int r = 0; r < TROWS; ++r) {
            const int j = t * TROWS + r;
            if (j < jStart) continue;
            const float u1 = tile[wv][buf][lane][r][0];
            const float u2 = tile[wv][buf][lane][r][1];

            const float s = (c - mo) * inv_so;
            const float ib = sigm(fmaf(s, wyum, byum) + u1 * wyum_u);
            const float oo = oo1 * sigm(fmaf(s, wyom, byom));
            const float oogw = oogw1 * sigm(fmaf(s, wgw, bgw));
            const float ol = ol1 * sigm(fmaf((u2 - ML) * inv_SL, wylm, bylm));
            const bool pos = (c > 0.0f);
            const float ratio = u2 / (pos ? c : 1.0f);
            const float olc = pos ? fminf(ol, ratio) : ol;
            const float f = 1.0f - oo - oogw - olc;
            const float c1 = f * c + (1.0f - ib) * u1;

            const int oi = j - WARM;
            if (kValid && oi >= 0 && oi < outLen) {
                const int row = time_lag + k * OUT_LEN + oi;
                const float bp = ib * u1;
                out[row] = fmaf(oo, c, bp);                  // h_n
                out[(size_t)1 * B + row] = c;                // c_n
                out[(size_t)2 * B + row] = ol * c;           // l_n
                out[(size_t)3 * B + row] = olc * c;          // lc_n
                out[(size_t)4 * B + row] = bp;               // bp_n
                out[(size_t)5 * B + row] = oogw * c;         // gw_n
                out[(size_t)6 * B + row] = ib;               // g_ib
                out[(size_t)7 * B + row] = oo;               // g_oo
                out[(size_t)8 * B + row] = ol;               // g_ol
                out[(size_t)9 * B + row] = olc;              // g_olc
                out[(size_t)10 * B + row] = f;               // g_f
                out[(size_t)11 * B + row] = oogw;            // g_oogw
                out[(size_t)12 * B + 2 * row] = fmaf(oo, c, bp); // h_nout[:,0]
                out[(size_t)12 * B + 2 * row + 1] = stdv;        // h_nout[:,1]
                out[(size_t)14 * B + row] = stdv;                // obs_std
            }
            c = c1;
        }
        buf ^= 1;
    }
}

// ---------------- host launcher ----------------

extern "C" void kernel_launch(void* const* d_in, const int* in_sizes, int n_in,
                              void* d_out, int out_size, void* d_ws,
                              size_t ws_size, hipStream_t stream) {
    (void)n_in; (void)out_size; (void)ws_size;
    const float* x = (const float*)d_in[0];
    const int* time_lag = (const int*)d_in[2];
    const float* y_obs = (const float*)d_in[3];
    const float* p_mean = (const float*)d_in[4];
    const float* p_std = (const float*)d_in[5];
    const float* w_r_yom = (const float*)d_in[6];
    const float* w_r_yom_gw = (const float*)d_in[7];
    const float* w_r_ylm = (const float*)d_in[8];
    const float* w_r_yfm = (const float*)d_in[9];
    const float* b0_yom = (const float*)d_in[10];
    const float* w1_yom = (const float*)d_in[11];
    const float* b0_gw = (const float*)d_in[12];
    const float* w1_gw = (const float*)d_in[13];
    const float* b0_ylm = (const float*)d_in[14];
    const float* w2_ylm = (const float*)d_in[15];
    const float* w1_yum = (const float*)d_in[16];
    const float* b0_yum = (const float*)d_in[17];
    float* out = (float*)d_out;
    double* acc = (double*)d_ws;

    const int B = in_sizes[3];                      // y_obs rows
    int n1 = (TRAIN < B) ? TRAIN : B;
    if (n1 < SPIN) n1 = SPIN;

    ws_init_kernel<<<1, 1, 0, stream>>>(acc);
    stats_kernel<<<256, TPB, 0, stream>>>(y_obs, SPIN, n1, acc);
    finalize_kernel<<<1, 1, 0, stream>>>(acc, n1 - SPIN);
    zerofill_kernel<<<256, TPB, 0, stream>>>(out, time_lag, B);

    const int chunkMax = (B + OUT_LEN - 1) / OUT_LEN;
    const int blocks = (chunkMax + TPB - 1) / TPB;
    mcpbrnn_scan_kernel<<<blocks, TPB, 0, stream>>>(
        x, time_lag, p_mean, p_std, w_r_yom, w_r_yom_gw, w_r_ylm, w_r_yfm,
        b0_yom, w1_yom, b0_gw, w1_gw, b0_ylm, w2_ylm, w1_yum, b0_yum,
        acc, out, B);
}